// RESK2_40956808135035
// MI455X (gfx1250) — compile-verified
//
#include <hip/hip_runtime.h>
#include <hip/hip_bf16.h>
#include <math.h>

// ---------------------------------------------------------------------------
// RESK2 4-layer GCN for MI455X (gfx1250, wave32).
//  - GEMM: V_WMMA_F32_16X16X4_F32, one 16x16 output tile per wave.
//    N % 16 == 0 (100000 = 6250*16) so row guards are unnecessary; the
//    Fout=40 layer uses clamped addressing + select (no exec-predicated loads).
//  - Scatter: one wave per edge, lane owns float4 of features, L2-resident
//    global_atomic_add_f32; global_prefetch_b8 for the next gather row.
// ---------------------------------------------------------------------------

typedef __attribute__((ext_vector_type(2))) float v2f;
typedef __attribute__((ext_vector_type(8))) float v8f;

#define WAVE 32

// H[nrows x Fout] = X[nrows x K] @ W[K x Fout] + bias
// One wave computes one 16x16 tile of H. K % 4 == 0, nrows % 16 == 0.
// FULL: Fout % 16 == 0 (no column guards at all).
template <bool FULL>
__global__ void gcn_gemm_wmma_f32(const float* __restrict__ X,
                                  const float* __restrict__ W,
                                  const float* __restrict__ bias,
                                  float* __restrict__ H,
                                  int K, int Fout, int colTiles,
                                  int totalTiles) {
    const int tileId = blockIdx.x * (blockDim.x >> 5) + (threadIdx.x >> 5);
    if (tileId >= totalTiles) return;   // wave-uniform

    const int lane = threadIdx.x & 31;
    const int m    = lane & 15;   // A-row / B-col within tile
    const int half = lane >> 4;   // K-half selector

    const int rowTile = tileId / colTiles;
    const int colTile = tileId - rowTile * colTiles;
    const int row0 = rowTile << 4;
    const int col0 = colTile << 4;

    const int  colB  = col0 + m;
    const bool colOK = FULL || (colB < Fout);
    const int  colC  = FULL ? colB : (colOK ? colB : (Fout - 1)); // clamped

    // Hoisted base pointers; inner loop is load+wmma only.
    const float* __restrict__ xrow = X + (size_t)(row0 + m) * (size_t)K + (half << 1);
    const float* __restrict__ wcol = W + (size_t)(half << 1) * (size_t)Fout + colC;

    v8f c = {0.f, 0.f, 0.f, 0.f, 0.f, 0.f, 0.f, 0.f};

    #pragma unroll 4
    for (int k = 0; k < K; k += 4) {
        const float2 av = *(const float2*)(xrow + k);                 // b64 load
        const float  bx = wcol[(size_t)k * (size_t)Fout];             // b32 load
        const float  by = wcol[(size_t)(k + 1) * (size_t)Fout];       // b32 load
        v2f a; a.x = av.x; a.y = av.y;
        v2f b;
        b.x = FULL ? bx : (colOK ? bx : 0.f);   // v_cndmask, not exec-branch
        b.y = FULL ? by : (colOK ? by : 0.f);
        // (neg_a, A, neg_b, B, c_mod, C, reuse_a, reuse_b)
        c = __builtin_amdgcn_wmma_f32_16x16x4_f32(false, a, false, b,
                                                  (short)0, c, false, false);
    }

    const float bv = FULL ? bias[colB] : (colOK ? bias[colC] : 0.f);
    // D layout: VGPR v -> row = row0 + v + 8*half, col = col0 + (lane&15)
    float* __restrict__ hout = H + (size_t)(row0 + (half << 3)) * (size_t)Fout + colB;
    #pragma unroll
    for (int v = 0; v < 8; ++v) {
        if (FULL || colOK) {
            hout[(size_t)v * (size_t)Fout] = c[v] + bv;
        }
    }
}

__global__ void zero_f32(float* __restrict__ p, long n) {
    long i = (long)blockIdx.x * blockDim.x + threadIdx.x;
    const long stride = (long)gridDim.x * blockDim.x;
    for (; i < n; i += stride) p[i] = 0.f;
}

// out[tgt[e]][:] += H[src[e]][:] * ew[e], F = 128. One wave per edge.
__global__ void scatter128(const float* __restrict__ H,
                           const int* __restrict__ src,
                           const int* __restrict__ tgt,
                           const float* __restrict__ ew,
                           float* __restrict__ out, int E) {
    const int wavesPerBlock = blockDim.x >> 5;
    const int lane = threadIdx.x & 31;
    int e = blockIdx.x * wavesPerBlock + (threadIdx.x >> 5);
    const int wstride = gridDim.x * wavesPerBlock;

    for (; e < E; e += wstride) {
        // prefetch next iteration's gather row (random access -> hide latency)
        const int en = e + wstride;
        if (en < E) {
            const int sn = src[en];
            __builtin_prefetch(H + (size_t)sn * 128 + (size_t)lane * 4, 0, 0);
        }
        const int   s = src[e];
        const int   t = tgt[e];
        const float w = ew[e];
        const float4 v = *(const float4*)(H + (size_t)s * 128 + (size_t)lane * 4);
        float* o = out + (size_t)t * 128 + (size_t)lane * 4;
        atomicAdd(o + 0, v.x * w);
        atomicAdd(o + 1, v.y * w);
        atomicAdd(o + 2, v.z * w);
        atomicAdd(o + 3, v.w * w);
    }
}

// F = 40 scatter, flat (edge, feature) grid-stride.
__global__ void scatter40(const float* __restrict__ H,
                          const int* __restrict__ src,
                          const int* __restrict__ tgt,
                          const float* __restrict__ ew,
                          float* __restrict__ out, int E) {
    const long total = (long)E * 40;
    long i = (long)blockIdx.x * blockDim.x + threadIdx.x;
    const long stride = (long)gridDim.x * blockDim.x;
    for (; i < total; i += stride) {
        const int e = (int)(i / 40);
        const int f = (int)(i - (long)e * 40);
        const float v = H[(size_t)src[e] * 40 + f] * ew[e];
        atomicAdd(out + (size_t)tgt[e] * 40 + f, v);
    }
}

__global__ void relu_copy(const float* __restrict__ in, float* __restrict__ out, long n) {
    long i = (long)blockIdx.x * blockDim.x + threadIdx.x;
    const long stride = (long)gridDim.x * blockDim.x;
    for (; i < n; i += stride) out[i] = fmaxf(in[i], 0.f);
}

// out = relu(in) + res
__global__ void relu_add(const float* __restrict__ in, const float* __restrict__ res,
                         float* __restrict__ out, long n) {
    long i = (long)blockIdx.x * blockDim.x + threadIdx.x;
    const long stride = (long)gridDim.x * blockDim.x;
    for (; i < n; i += stride) out[i] = fmaxf(in[i], 0.f) + res[i];
}

// log_softmax over 40 values per row; one wave per row.
__global__ void log_softmax40(const float* __restrict__ in, float* __restrict__ out,
                              int nrows) {
    const int row = blockIdx.x * (blockDim.x >> 5) + (threadIdx.x >> 5);
    if (row >= nrows) return;   // wave-uniform
    const int lane = threadIdx.x & 31;
    const bool has2 = lane < 8;

    const float* r = in + (size_t)row * 40;
    const float v0 = r[lane];
    const float v1 = has2 ? r[lane + 32] : -INFINITY;

    float m = fmaxf(v0, v1);
    #pragma unroll
    for (int off = 16; off > 0; off >>= 1)
        m = fmaxf(m, __shfl_xor(m, off, WAVE));

    float s = expf(v0 - m) + (has2 ? expf(v1 - m) : 0.f);
    #pragma unroll
    for (int off = 16; off > 0; off >>= 1)
        s += __shfl_xor(s, off, WAVE);

    const float l = m + logf(s);
    float* o = out + (size_t)row * 40;
    o[lane] = v0 - l;
    if (has2) o[lane + 32] = v1 - l;
}

// ---------------------------------------------------------------------------

static inline int cdiv(long a, long b) { return (int)((a + b - 1) / b); }

extern "C" void kernel_launch(void* const* d_in, const int* in_sizes, int n_in,
                              void* d_out, int out_size, void* d_ws, size_t ws_size,
                              hipStream_t stream) {
    (void)n_in; (void)ws_size; (void)out_size; (void)in_sizes;

    const int N = 100000, E = 1600000;
    const int F0 = 256, F = 128, C = 40;

    const float* x   = (const float*)d_in[0];
    const int*   src = (const int*)  d_in[1];
    const int*   tgt = (const int*)  d_in[2];
    const float* ew  = (const float*)d_in[3];
    const float* W0  = (const float*)d_in[4];
    const float* b0  = (const float*)d_in[5];
    const float* W1  = (const float*)d_in[6];
    const float* b1  = (const float*)d_in[7];
    const float* W2  = (const float*)d_in[8];
    const float* b2  = (const float*)d_in[9];
    const float* W3  = (const float*)d_in[10];
    const float* b3  = (const float*)d_in[11];
    float* out = (float*)d_out;

    // Workspace: 4 buffers of N*128 floats (~205 MB total).
    float* h   = (float*)d_ws;          // gemm output (also reused F=40)
    float* agg = h   + (size_t)N * F;   // scatter accumulator (also F=40 logits)
    float* x0  = agg + (size_t)N * F;   // residual
    float* xc  = x0  + (size_t)N * F;   // current activation

    const long nf   = (long)N * F;
    const long nc   = (long)N * C;
    const int  rowT = N / 16;           // 6250, exact

    const dim3 blk(256);
    const int gemmBlocks128 = cdiv((long)rowT * 8, 8);   // 8 col tiles, 8 waves/blk
    const int gemmBlocks40  = cdiv((long)rowT * 3, 8);   // 3 col tiles
    const int zeroBlocks    = cdiv(nf, 256);
    const int scatBlocks    = cdiv(E, 8);                // 8 waves/block, 1 edge/wave
    const int eltBlocks     = cdiv(nf, 256);
    const int smBlocks      = cdiv(N, 8);

    // ---- Layer 0: h = x@W0+b0 ; agg = scatter ; x0 = relu(agg) ----
    gcn_gemm_wmma_f32<true><<<gemmBlocks128, blk, 0, stream>>>(x, W0, b0, h, F0, F, 8, rowT * 8);
    zero_f32<<<zeroBlocks, blk, 0, stream>>>(agg, nf);
    scatter128<<<scatBlocks, blk, 0, stream>>>(h, src, tgt, ew, agg, E);
    relu_copy<<<eltBlocks, blk, 0, stream>>>(agg, x0, nf);

    // ---- Layer 1: h = x0@W1+b1 ; xc = relu(scatter) ----
    gcn_gemm_wmma_f32<true><<<gemmBlocks128, blk, 0, stream>>>(x0, W1, b1, h, F, F, 8, rowT * 8);
    zero_f32<<<zeroBlocks, blk, 0, stream>>>(agg, nf);
    scatter128<<<scatBlocks, blk, 0, stream>>>(h, src, tgt, ew, agg, E);
    relu_copy<<<eltBlocks, blk, 0, stream>>>(agg, xc, nf);

    // ---- Layer 2: h = xc@W2+b2 ; xc = relu(scatter) + x0 ----
    gcn_gemm_wmma_f32<true><<<gemmBlocks128, blk, 0, stream>>>(xc, W2, b2, h, F, F, 8, rowT * 8);
    zero_f32<<<zeroBlocks, blk, 0, stream>>>(agg, nf);
    scatter128<<<scatBlocks, blk, 0, stream>>>(h, src, tgt, ew, agg, E);
    relu_add<<<eltBlocks, blk, 0, stream>>>(agg, x0, xc, nf);

    // ---- Layer 3: h(:, :40) = xc@W3+b3 ; logits = scatter40 ; log_softmax ----
    gcn_gemm_wmma_f32<false><<<gemmBlocks40, blk, 0, stream>>>(xc, W3, b3, h, F, C, 3, rowT * 3);
    zero_f32<<<cdiv(nc, 256), blk, 0, stream>>>(agg, nc);
    {
        long sb = cdiv((long)E * C, 256);
        if (sb > 65535) sb = 65535;
        scatter40<<<(int)sb, blk, 0, stream>>>(h, src, tgt, ew, agg, E);
    }
    log_softmax40<<<smBlocks, blk, 0, stream>>>(agg, out, N);
}